// KECGMultiHeadGraphAttention_34394098106411
// MI455X (gfx1250) — compile-verified
//
#include <hip/hip_runtime.h>
#include <hip/hip_bf16.h>

typedef __attribute__((ext_vector_type(2))) float v2f;
typedef __attribute__((ext_vector_type(8))) float v8f;

#define N_NODES   100000
#define N_EDGES   1600000
#define N_HEAD    2
#define F_OUT     64
#define NEG_SLOPE 0.2f

#define N_TILES   (N_NODES / 16)               // 6250 exactly
#define OUT_ELEMS (N_HEAD * N_NODES * F_OUT)   // 12.8M floats
#define RS_ELEMS  (N_HEAD * N_NODES)           // 200K floats

// ---------------------------------------------------------------------------
// Kernel 0: zero the accumulators (d_out used directly as agg, rowsum in ws)
// ---------------------------------------------------------------------------
__global__ void gat_zero(float4* __restrict__ out4, float4* __restrict__ rs4) {
    size_t i = (size_t)blockIdx.x * blockDim.x + threadIdx.x;
    size_t stride = (size_t)gridDim.x * blockDim.x;
    const size_t n_out4 = OUT_ELEMS / 4;
    const size_t n_rs4  = RS_ELEMS / 4;
    float4 z; z.x = 0.f; z.y = 0.f; z.z = 0.f; z.w = 0.f;
    for (size_t j = i; j < n_out4; j += stride) out4[j] = z;
    for (size_t j = i; j < n_rs4;  j += stride) rs4[j]  = z;
}

// ---------------------------------------------------------------------------
// Kernel 1: per-node scores via V_WMMA_F32_16X16X4_F32.
// S(100000x4) = X(100000x64) @ C(64x4), C[:,n] = w[n>>1] * a[n>>1, (n&1)*F..]
// S stored per node as float4: {s_src_h0, s_dst_h0, s_src_h1, s_dst_h1}.
// One wave per 16-node tile, 16 WMMA steps of K=4 accumulate F=64.
// B fragments are tile-invariant (hoisted to 16 x v2f registers); coef LDS
// is padded to 16 columns (cols 4..15 = 0) so the B load is unconditional.
// D stores use a single base pointer + immediate offsets.
// ---------------------------------------------------------------------------
__global__ void gat_scores_wmma(const float* __restrict__ x,
                                const float* __restrict__ wmat,     // (H,1,F) flat: 128
                                const float* __restrict__ avec,     // (H,2F,1) flat: 256
                                float* __restrict__ S) {            // N*4 floats
    __shared__ float coef[16 * F_OUT];  // coef[n*64 + f]; n>=4 -> 0
    for (int t = threadIdx.x; t < 16 * F_OUT; t += blockDim.x) {
        int n = t >> 6, f = t & 63;
        float v = 0.f;
        if (n < 4) {
            int h = n >> 1, part = n & 1;
            v = wmat[h * F_OUT + f] * avec[h * 2 * F_OUT + part * F_OUT + f];
        }
        coef[t] = v;
    }
    __syncthreads();

    const int lane  = threadIdx.x & 31;
    const int m     = lane & 15;        // row (A/M) or column (B,D/N)
    const int kh    = lane >> 4;        // K half: lanes 16-31 hold K+2

    // Hoisted, unconditional B fragments: vgpr v holds row K = kh*2+v, col = m
    v2f bfrag[16];
    #pragma unroll
    for (int kk = 0; kk < 16; ++kk)
        bfrag[kk] = *(const v2f*)&coef[m * F_OUT + kk * 4 + kh * 2];

    const int wavesPerBlk = blockDim.x >> 5;
    const int waveG = blockIdx.x * wavesPerBlk + (threadIdx.x >> 5);
    const int nWaves = gridDim.x * wavesPerBlk;

    for (int tile = waveG; tile < N_TILES; tile += nWaves) {
        const int node0 = tile * 16;
        const float* xrow = x + (size_t)(node0 + m) * F_OUT;

        v8f acc = {};
        #pragma unroll
        for (int kk = 0; kk < 16; ++kk) {
            // A 16x4 f32: vgpr v holds K = kh*2 + v -> contiguous float2
            v2f a = *(const v2f*)(xrow + kk * 4 + kh * 2);
            acc = __builtin_amdgcn_wmma_f32_16x16x4_f32(
                false, a, false, bfrag[kk], (short)0, acc, false, false);
        }

        // D 16x16: vgpr j -> row node0+j+kh*8, col = m. Stores are a fixed
        // 16-byte stride from one base -> immediate-offset global stores.
        if (m < 4) {
            float* sp = S + (size_t)(node0 + kh * 8) * 4 + m;
            #pragma unroll
            for (int j = 0; j < 8; ++j)
                sp[j * 4] = acc[j];
        }
    }
}

// ---------------------------------------------------------------------------
// Kernel 2: edge scatter. One wave32 per edge; lane handles 2 features.
//   logit_h = s_src_h[src] + s_dst_h[dst]
//   ee_h    = exp(-leaky_relu(logit_h))
//   rowsum[h,src] += ee_h ;  agg[h,src,f] += ee_h * w[h,f] * x[dst,f]
// ---------------------------------------------------------------------------
__global__ void gat_edge(const float* __restrict__ x,
                         const int* __restrict__ edge,     // (2,E) flat, int32
                         const float* __restrict__ wmat,   // 128 floats
                         const float4* __restrict__ S4,
                         float* __restrict__ agg,          // (H,N,F) = d_out
                         float* __restrict__ rowsum) {     // (H,N)
    __shared__ float wsh[N_HEAD * F_OUT];
    if (threadIdx.x < N_HEAD * F_OUT) wsh[threadIdx.x] = wmat[threadIdx.x];
    __syncthreads();

    const int lane = threadIdx.x & 31;
    const long waveG  = (long)blockIdx.x * (blockDim.x >> 5) + (threadIdx.x >> 5);
    const long nWaves = (long)gridDim.x * (blockDim.x >> 5);

    for (long e = waveG; e < N_EDGES; e += nWaves) {
        const int src = edge[e];
        const int dst = edge[N_EDGES + e];

        // broadcast loads (same address across wave) -> single request
        float4 ss = S4[src];
        float4 sd = S4[dst];
        float l0 = ss.x + sd.y;
        float l1 = ss.z + sd.w;
        l0 = (l0 > 0.f) ? l0 : NEG_SLOPE * l0;
        l1 = (l1 > 0.f) ? l1 : NEG_SLOPE * l1;
        const float ee0 = __expf(-l0);
        const float ee1 = __expf(-l1);

        if (lane < 2) {
            unsafeAtomicAdd(&rowsum[lane * N_NODES + src], lane ? ee1 : ee0);
        }

        const int f0 = lane * 2;
        const float2 xv = *(const float2*)(x + (size_t)dst * F_OUT + f0);

        float* a0 = agg + (size_t)src * F_OUT + f0;
        float* a1 = agg + (size_t)N_NODES * F_OUT + (size_t)src * F_OUT + f0;
        unsafeAtomicAdd(a0,     ee0 * wsh[f0]     * xv.x);
        unsafeAtomicAdd(a0 + 1, ee0 * wsh[f0 + 1] * xv.y);
        unsafeAtomicAdd(a1,     ee1 * wsh[F_OUT + f0]     * xv.x);
        unsafeAtomicAdd(a1 + 1, ee1 * wsh[F_OUT + f0 + 1] * xv.y);
    }
}

// ---------------------------------------------------------------------------
// Kernel 3: out = agg / rowsum   (in place on d_out, float4 streaming)
// ---------------------------------------------------------------------------
__global__ void gat_div(float4* __restrict__ out4, const float* __restrict__ rowsum) {
    size_t i = (size_t)blockIdx.x * blockDim.x + threadIdx.x;
    size_t stride = (size_t)gridDim.x * blockDim.x;
    const size_t n4 = OUT_ELEMS / 4;               // 3.2M float4
    for (size_t j = i; j < n4; j += stride) {
        float r = rowsum[j >> 4];                   // 16 float4 per (h,n)
        float4 v = out4[j];
        v.x /= r; v.y /= r; v.z /= r; v.w /= r;
        out4[j] = v;
    }
}

// ---------------------------------------------------------------------------
extern "C" void kernel_launch(void* const* d_in, const int* in_sizes, int n_in,
                              void* d_out, int out_size, void* d_ws, size_t ws_size,
                              hipStream_t stream) {
    const float* x    = (const float*)d_in[0];
    const int*   edge = (const int*)d_in[1];
    const float* wmat = (const float*)d_in[2];
    const float* avec = (const float*)d_in[3];

    float* agg = (float*)d_out;
    float* S   = (float*)d_ws;                     // N*4 floats = 1.6 MB
    float* rowsum = S + (size_t)N_NODES * 4;       // H*N floats = 0.8 MB

    // 0) zero accumulators
    gat_zero<<<2048, 256, 0, stream>>>((float4*)agg, (float4*)rowsum);

    // 1) per-node scores via WMMA: 6250 tiles, 8 waves per 256-thread block
    const int sBlocks = (N_TILES + 7) / 8;         // 782
    gat_scores_wmma<<<sBlocks, 256, 0, stream>>>(x, wmat, avec, S);

    // 2) edge scatter: one wave per edge
    const int eBlocks = N_EDGES / 8;               // 200000 blocks of 8 waves
    gat_edge<<<eBlocks, 256, 0, stream>>>(x, edge, wmat, (const float4*)S, agg, rowsum);

    // 3) normalize
    gat_div<<<4096, 256, 0, stream>>>((float4*)agg, rowsum);
}